// SentenceEmbedderEntailment_87093346828777
// MI455X (gfx1250) — compile-verified
//
#include <hip/hip_runtime.h>
#include <math.h>

typedef __attribute__((ext_vector_type(2))) float v2f;
typedef __attribute__((ext_vector_type(8))) float v8f;

constexpr int EMB    = 100;   // embed size (fixed in reference)
constexpr int NSLICE = 128;   // partial-result slices to de-contend atomics

// ---------------------------------------------------------------------------
// Workspace layout (floats):
//   u    : [2 cubes][2 sentences][NV][EMB]   (per-verb accumulated arg vectors)
//   P    : [NSLICE][2 sentences][EMB]        (partial embedding accumulators)
//   zpad : [128]                             (zeros; B-fragment source for lanes N>=2)
// ---------------------------------------------------------------------------

__global__ void zero_kernel(float* p, long long n) {
    long long i = (long long)blockIdx.x * blockDim.x + threadIdx.x;
    long long stride = (long long)gridDim.x * blockDim.x;
    for (; i < n; i += stride) p[i] = 0.f;
}

// Scatter: u[cube][s][k] += noun_emb[arg].  8 edge lists x NV edges.
__global__ void scatter_kernel(const int* __restrict__ vs1, const int* __restrict__ vo1,
                               const int* __restrict__ vt1, const int* __restrict__ vs2,
                               const int* __restrict__ vo2, const int* __restrict__ vt2,
                               const float* __restrict__ noun, float* __restrict__ u, int NV) {
    int op = blockIdx.x / NV;       // 0..7
    int n  = blockIdx.x % NV;
    int t  = threadIdx.x;
    int s   = op >> 2;              // sentence
    int sub = op & 3;
    const int* vs = s ? vs2 : vs1;
    const int* vo = s ? vo2 : vo1;
    const int* vt = s ? vt2 : vt1;
    int k, a, cube;
    switch (sub) {
        case 0:  k = vs[2*n];  a = vs[2*n+1];  cube = 0; break;   // subj verb
        case 1:  k = vo[2*n];  a = vo[2*n+1];  cube = 1; break;   // obj verb
        case 2:  k = vt[3*n];  a = vt[3*n+1];  cube = 0; break;   // trans, subj arg
        default: k = vt[3*n];  a = vt[3*n+2];  cube = 1; break;   // trans, obj arg
    }
    if (t < EMB) {
        float val = noun[(size_t)a * EMB + t];
        atomicAdd(&u[((size_t)cube * 2 + s) * (size_t)NV * EMB + (size_t)k * EMB + t], val);
    }
}

// Word-sum: block-local register accumulation, one atomic per (block, dim).
__global__ void wordsum_kernel(const int* __restrict__ w1, const int* __restrict__ w2,
                               const float* __restrict__ noun, float* __restrict__ P,
                               int NW, int blocksPerS) {
    int s = blockIdx.x / blocksPerS;
    int b = blockIdx.x % blocksPerS;
    const int* w = s ? w2 : w1;
    int t = threadIdx.x;
    int chunk = (NW + blocksPerS - 1) / blocksPerS;
    int w0 = b * chunk;
    int w1e = w0 + chunk; if (w1e > NW) w1e = NW;
    if (t < EMB) {
        float acc = 0.f;
        for (int i = w0; i < w1e; i++) acc += noun[(size_t)w[i] * EMB + t];
        int slice = blockIdx.x & (NSLICE - 1);
        atomicAdd(&P[(size_t)slice * 2 * EMB + s * EMB + t], acc);
    }
}

// Bandwidth-dominant pass: emb_s[i] += sum_k cube[k,i,:] . U[cube][s][k][:]
// via V_WMMA_F32_16X16X4_F32, branch-free inner loop, ping-pong prefetch.
__global__ __launch_bounds__(256) void verb_wmma_kernel(
        const float* __restrict__ subjc, const float* __restrict__ objc,
        const float* __restrict__ u, float* __restrict__ P,
        const float* __restrict__ zpad,
        int NV, int wavesPerCube, int KPW) {
    int wave = blockIdx.x * (blockDim.x >> 5) + (threadIdx.x >> 5);
    if (wave >= 2 * wavesPerCube) return;             // wave-uniform exit
    int cubeSel = (wave >= wavesPerCube) ? 1 : 0;
    int kBase   = (wave - cubeSel * wavesPerCube) * KPW;
    const float* cube = cubeSel ? objc : subjc;
    const float* U0 = u + (size_t)cubeSel * 2 * NV * EMB;   // sentence 1
    const float* U1 = U0 + (size_t)NV * EMB;                // sentence 2

    int lane = threadIdx.x & 31;
    int m    = lane & 15;      // A: row M within tile; B/D: column N
    int half = lane >> 4;      // selects K pair (A/B) / +8 rows (D)
    int koff = 2 * half;       // this lane's K-pair offset within a 4-chunk
    int i6   = 96 + m; if (i6 > EMB - 1) i6 = EMB - 1;  // clamped row for tile 6

    v8f cacc[7];
    #pragma unroll
    for (int t = 0; t < 7; t++)
        #pragma unroll
        for (int v = 0; v < 8; v++) cacc[t][v] = 0.f;

#define LOADA(dst, base6_, rb_, j_)                                              \
    {   _Pragma("unroll")                                                        \
        for (int t = 0; t < 6; t++)                                              \
            dst[t] = __builtin_nontemporal_load(                                 \
                         (const v2f*)((rb_) + t * 16 * EMB + (j_)));             \
        dst[6] = __builtin_nontemporal_load((const v2f*)((base6_) + (j_)));      \
    }

#define WMMA7(af_, bf_)                                                          \
    {   _Pragma("unroll")                                                        \
        for (int t = 0; t < 7; t++)                                              \
            cacc[t] = __builtin_amdgcn_wmma_f32_16x16x4_f32(                     \
                false, af_[t], false, bf_, (short)0, cacc[t], false, false);     \
    }

    for (int kk = 0; kk < KPW; kk++) {
        int k = kBase + kk;
        if (k >= NV) break;
        const float* A  = cube + (size_t)k * EMB * EMB;
        // branch-free per-lane B source: lane N=0 -> U(sent1,k), N=1 -> U(sent2,k),
        // N>=2 -> zero pad (v_cndmask on pointers, no EXEC divergence)
        const float* bsel = (m == 0) ? (U0 + (size_t)k * EMB)
                          : (m == 1) ? (U1 + (size_t)k * EMB)
                          : zpad;
        bsel += koff;
        const float* rb = A + m * EMB + koff;        // tiles 0..5 base (this lane's row)
        const float* r6 = A + i6 * EMB + koff;       // tile 6 (row clamped; extras discarded)

        v2f a0[7], a1[7], b0, b1;
        // prologue: chunk j=0
        b0 = *(const v2f*)(bsel);
        LOADA(a0, r6, rb, 0)
        int j = 0;
        for (int it = 0; it < 12; ++it) {            // chunks j, j+4 ; j = 0,8,...,88
            b1 = *(const v2f*)(bsel + j + 4);
            LOADA(a1, r6, rb, j + 4)
            WMMA7(a0, b0)
            b0 = *(const v2f*)(bsel + j + 8);        // j+8 <= 96
            LOADA(a0, r6, rb, j + 8)
            WMMA7(a1, b1)
            j += 8;
        }
        WMMA7(a0, b0)                                // final chunk j=96
    }
#undef LOADA
#undef WMMA7

    // D layout: VGPR v, lanes 0-15 -> (M=v, N=lane), lanes 16-31 -> (M=v+8, N=lane-16).
    // Only N=0,1 carry real data.
    int slice = wave & (NSLICE - 1);
    if (m < 2) {
        float* Pd = P + (size_t)slice * 2 * EMB + m * EMB;
        #pragma unroll
        for (int t = 0; t < 7; t++)
            #pragma unroll
            for (int v = 0; v < 8; v++) {
                int i = 16 * t + v + 8 * half;
                if (i < EMB) atomicAdd(&Pd[i], cacc[t][v]);
            }
    }
}

// Final: reduce slices -> emb1/emb2, build vec_dist, MLP, log-softmax.
__global__ __launch_bounds__(256) void finalize_kernel(
        const float* __restrict__ P, const float* __restrict__ W_h,
        const float* __restrict__ b_h, const float* __restrict__ W_o,
        const float* __restrict__ b_o, float* __restrict__ out, int hidden) {
    __shared__ float emb[2 * EMB];
    __shared__ float vec[4 * EMB];
    __shared__ float hbuf[512];
    __shared__ float logits[3];
    int t = threadIdx.x;
    for (int idx = t; idx < 2 * EMB; idx += blockDim.x) {
        float s = 0.f;
        for (int sl = 0; sl < NSLICE; sl++) s += P[(size_t)sl * 2 * EMB + idx];
        emb[idx] = s;
    }
    __syncthreads();
    for (int i = t; i < EMB; i += blockDim.x) {
        float e1 = emb[i], e2 = emb[EMB + i];
        vec[i] = e1; vec[EMB + i] = e2;
        vec[2 * EMB + i] = e1 * e2;
        vec[3 * EMB + i] = fabsf(e1 - e2);
    }
    __syncthreads();
    for (int j = t; j < hidden; j += blockDim.x) {
        float acc = b_h[j];
        const float* w = W_h + (size_t)j * 4 * EMB;
        for (int ci = 0; ci < 4 * EMB; ci++) acc += w[ci] * vec[ci];
        hbuf[j] = 1.f / (1.f + expf(-acc));
    }
    __syncthreads();
    if (t < 3) {
        float acc = b_o[t];
        const float* w = W_o + (size_t)t * hidden;
        for (int j = 0; j < hidden; j++) acc += w[j] * hbuf[j];
        logits[t] = acc;
    }
    __syncthreads();
    if (t == 0) {
        float mx = fmaxf(logits[0], fmaxf(logits[1], logits[2]));
        float lse = mx + logf(expf(logits[0] - mx) + expf(logits[1] - mx) + expf(logits[2] - mx));
        out[0] = logits[0] - lse;
        out[1] = logits[1] - lse;
        out[2] = logits[2] - lse;
    }
}

extern "C" void kernel_launch(void* const* d_in, const int* in_sizes, int n_in,
                              void* d_out, int out_size, void* d_ws, size_t ws_size,
                              hipStream_t stream) {
    const int*   words1 = (const int*)d_in[0];
    const int*   vs1    = (const int*)d_in[1];
    const int*   vo1    = (const int*)d_in[2];
    const int*   vt1    = (const int*)d_in[3];
    const int*   words2 = (const int*)d_in[4];
    const int*   vs2    = (const int*)d_in[5];
    const int*   vo2    = (const int*)d_in[6];
    const int*   vt2    = (const int*)d_in[7];
    const float* noun   = (const float*)d_in[8];
    const float* subjc  = (const float*)d_in[9];
    const float* objc   = (const float*)d_in[10];
    const float* W_h    = (const float*)d_in[11];
    const float* b_h    = (const float*)d_in[12];
    const float* W_o    = (const float*)d_in[13];
    const float* b_o    = (const float*)d_in[14];

    int NW     = in_sizes[0];        // 50000 words per sentence
    int NV     = in_sizes[1] / 2;    // 5000 verbs per edge list
    int hidden = in_sizes[12];       // 300

    float* u = (float*)d_ws;
    size_t uFloats = (size_t)4 * NV * EMB;            // 2 cubes x 2 sentences
    float* P = u + uFloats;
    size_t pFloats = (size_t)NSLICE * 2 * EMB;
    float* zpad = P + pFloats;
    size_t zFloats = 128;
    long long totalZero = (long long)(uFloats + pFloats + zFloats);

    zero_kernel<<<512, 256, 0, stream>>>(u, totalZero);
    scatter_kernel<<<8 * NV, 128, 0, stream>>>(vs1, vo1, vt1, vs2, vo2, vt2, noun, u, NV);
    int blocksPerS = 128;
    wordsum_kernel<<<2 * blocksPerS, 128, 0, stream>>>(words1, words2, noun, P, NW, blocksPerS);

    const int KPW = 2;
    int wavesPerCube  = (NV + KPW - 1) / KPW;
    int totalWaves    = 2 * wavesPerCube;
    int wavesPerBlock = 256 / 32;
    int blocks        = (totalWaves + wavesPerBlock - 1) / wavesPerBlock;
    verb_wmma_kernel<<<blocks, 256, 0, stream>>>(subjc, objc, u, P, zpad, NV, wavesPerCube, KPW);

    finalize_kernel<<<1, 256, 0, stream>>>(P, W_h, b_h, W_o, b_o, (float*)d_out, hidden);
}